// Model_36807869726789
// MI455X (gfx1250) — compile-verified
//
#include <hip/hip_runtime.h>
#include <math.h>

#define B_   16
#define N_   4096
#define CH_  64
#define S_   1024
#define K_   32
#define EPS_ 1e-5f

typedef __attribute__((ext_vector_type(2))) float v2f;
typedef __attribute__((ext_vector_type(8))) float v8f;

// ---------------------------------------------------------------------------
// Kernel 1: farthest point sampling. One 1024-thread block per batch.
// Each thread owns 4 points in VGPRs; argmax via wave32 shuffles + LDS.
// Writes fps indices (workspace) and new_xyz (first region of d_out).
// ---------------------------------------------------------------------------
__global__ __launch_bounds__(1024) void fps_kernel(const float* __restrict__ xyz,
                                                   float* __restrict__ new_xyz,
                                                   int* __restrict__ fps_idx)
{
    const int b    = blockIdx.x;
    const int tid  = threadIdx.x;
    const int lane = tid & 31;
    const int wv   = tid >> 5;            // 32 waves

    __shared__ int   s_sel[S_];
    __shared__ float s_rv[32];
    __shared__ int   s_ri[32];
    __shared__ int   s_last;

    const float* bx = xyz + (size_t)b * N_ * 3;

    float px[4], py[4], pz[4], md[4];
#pragma unroll
    for (int j = 0; j < 4; ++j) {
        int n = tid + 1024 * j;
        px[j] = bx[n * 3 + 0];
        py[j] = bx[n * 3 + 1];
        pz[j] = bx[n * 3 + 2];
        md[j] = INFINITY;
    }

    int last = 0;
    for (int t = 0; t < S_; ++t) {
        if (tid == 0) s_sel[t] = last;
        const float lx = bx[last * 3 + 0];
        const float ly = bx[last * 3 + 1];
        const float lz = bx[last * 3 + 2];

        float bv = -1.0f; int bi = 0;
#pragma unroll
        for (int j = 0; j < 4; ++j) {
            float dx = px[j] - lx, dy = py[j] - ly, dz = pz[j] - lz;
            float d  = dx * dx + dy * dy + dz * dz;
            md[j] = fminf(md[j], d);
            if (md[j] > bv) { bv = md[j]; bi = tid + 1024 * j; }   // ascending n -> first-max kept
        }
        // wave32 argmax reduction (tie -> smaller index, like jnp.argmax)
#pragma unroll
        for (int off = 16; off >= 1; off >>= 1) {
            float ov = __shfl_xor(bv, off, 32);
            int   oi = __shfl_xor(bi, off, 32);
            if (ov > bv || (ov == bv && oi < bi)) { bv = ov; bi = oi; }
        }
        if (lane == 0) { s_rv[wv] = bv; s_ri[wv] = bi; }
        __syncthreads();
        if (tid < 32) {
            float rv = s_rv[lane]; int ri = s_ri[lane];
#pragma unroll
            for (int off = 16; off >= 1; off >>= 1) {
                float ov = __shfl_xor(rv, off, 32);
                int   oi = __shfl_xor(ri, off, 32);
                if (ov > rv || (ov == rv && oi < ri)) { rv = ov; ri = oi; }
            }
            if (tid == 0) s_last = ri;
        }
        __syncthreads();
        last = s_last;
    }

    const int sidx = s_sel[tid];
    fps_idx[(size_t)b * S_ + tid] = sidx;
    float* nx = new_xyz + ((size_t)b * S_ + tid) * 3;
    nx[0] = bx[sidx * 3 + 0];
    nx[1] = bx[sidx * 3 + 1];
    nx[2] = bx[sidx * 3 + 2];
}

// ---------------------------------------------------------------------------
// Kernel 2: pairwise distances via V_WMMA_F32_16X16X4_F32 + top-K selection.
// One 512-thread (16-wave) block per (batch, 16-sample tile). The full
// 16 x 4096 distance panel lives in LDS (256 KB of the WGP's 320 KB).
// A = -2 * sample coords (16x4, dim3 padded 0), B = candidate coords (4x16).
// Each wave then extracts the 32 nearest neighbors for one sample row.
// ---------------------------------------------------------------------------
__global__ __launch_bounds__(512) void knn_kernel(const float* __restrict__ xyz,
                                                  const float* __restrict__ new_xyz,
                                                  int* __restrict__ knn_idx)
{
    extern __shared__ float ldsDist[];            // 16 * 4096 floats = 256 KB
    const int b    = blockIdx.x >> 6;             // S_/16 = 64 tiles per batch
    const int s0   = (blockIdx.x & 63) << 4;
    const int tid  = threadIdx.x;
    const int lane = tid & 31;
    const int wv   = tid >> 5;                    // 0..15
    const int half = lane >> 4;                   // A/B K-pair select
    const int l15  = lane & 15;

    // A operand: row m = l15 (sample s0+m); lanes 0-15 hold K={0,1}, 16-31 K={2,3}
    const float* sp = new_xyz + ((size_t)b * S_ + s0 + l15) * 3;
    v2f av;
    if (half == 0) { av.x = -2.0f * sp[0]; av.y = -2.0f * sp[1]; }
    else           { av.x = -2.0f * sp[2]; av.y = 0.0f; }

    const float* bxyz = xyz + (size_t)b * N_ * 3;

    for (int i = 0; i < 16; ++i) {
        const int nbase = (wv + 16 * i) << 4;     // 256 chunks of 16 candidates
        const float* cp = bxyz + (size_t)(nbase + l15) * 3;
        const float cx = cp[0], cy = cp[1], cz = cp[2];
        v2f bv2;
        if (half == 0) { bv2.x = cx; bv2.y = cy; }
        else           { bv2.x = cz; bv2.y = 0.0f; }
        const float nn = cx * cx + cy * cy + cz * cz;   // |n|^2 (|s|^2 is order-invariant)

        v8f acc = {};
        acc = __builtin_amdgcn_wmma_f32_16x16x4_f32(
                  false, av, false, bv2, (short)0, acc, false, false);

        // D layout: VGPR r holds row (r + 8*half), col = nbase + l15
        const int mb  = half * 8;
        const int col = nbase + l15;
#pragma unroll
        for (int r = 0; r < 8; ++r)
            ldsDist[(mb + r) * N_ + col] = acc[r] + nn;
    }
    __syncthreads();

    // Selection: wave wv handles sample row wv; 32 rounds of min-extraction.
    float* row = ldsDist + wv * N_;
    const size_t obase = ((size_t)b * S_ + s0 + wv) * K_;
    for (int k = 0; k < K_; ++k) {
        float bd = INFINITY; int bi = 0;
        for (int j = 0; j < N_ / 32; ++j) {
            const int c = lane + 32 * j;          // ascending -> first-min kept
            const float v = row[c];
            if (v < bd) { bd = v; bi = c; }
        }
#pragma unroll
        for (int off = 16; off >= 1; off >>= 1) {
            float ov = __shfl_xor(bd, off, 32);
            int   oi = __shfl_xor(bi, off, 32);
            if (ov < bd || (ov == bd && oi < bi)) { bd = ov; bi = oi; }
        }
        if (lane == (bi & 31)) row[bi] = INFINITY;  // mask out winner
        if (lane == 0) knn_idx[obase + k] = bi;
    }
}

// ---------------------------------------------------------------------------
// Kernel 3: gather + per-group normalization + affine + concat with anchor.
// 256 threads = 4 samples x 64 channels; channel-contiguous -> coalesced.
// ---------------------------------------------------------------------------
__global__ __launch_bounds__(256) void group_kernel(const float* __restrict__ points,
                                                    const float* __restrict__ alpha,
                                                    const float* __restrict__ beta,
                                                    const int* __restrict__ fps_idx,
                                                    const int* __restrict__ knn_idx,
                                                    float* __restrict__ out_np)
{
    const int tid = threadIdx.x;
    const int c   = tid & 63;
    const int g   = blockIdx.x * 4 + (tid >> 6);   // flattened (b,s), 0..B*S-1
    const int b   = g >> 10;                       // S_ == 1024

    const float* bp = points + (size_t)b * N_ * CH_;
    const float anchor = bp[(size_t)fps_idx[g] * CH_ + c];
    const int* ki = knn_idx + (size_t)g * K_;

    float diff[K_];
    float sum = 0.0f, sumsq = 0.0f;
#pragma unroll
    for (int k = 0; k < K_; ++k) {
        const float v = bp[(size_t)ki[k] * CH_ + c];
        const float d = v - anchor;
        diff[k] = d; sum += d; sumsq += d * d;
    }
    const float mu   = sum * (1.0f / K_);
    const float var  = (sumsq - (float)K_ * mu * mu) * (1.0f / (K_ - 1));
    const float stdv = sqrtf(fmaxf(var, 0.0f));
    const float inv  = 1.0f / (stdv + EPS_);
    const float al = alpha[c], be = beta[c];

    float* orow = out_np + (size_t)g * K_ * (2 * CH_);
#pragma unroll
    for (int k = 0; k < K_; ++k) {
        orow[k * (2 * CH_) + c]       = al * (diff[k] * inv) + be;
        orow[k * (2 * CH_) + CH_ + c] = anchor;
    }
}

// ---------------------------------------------------------------------------
extern "C" void kernel_launch(void* const* d_in, const int* in_sizes, int n_in,
                              void* d_out, int out_size, void* d_ws, size_t ws_size,
                              hipStream_t stream)
{
    const float* xyz    = (const float*)d_in[0];   // (B,N,3)
    const float* points = (const float*)d_in[1];   // (B,N,CH)
    const float* alpha  = (const float*)d_in[2];   // (1,1,1,CH)
    const float* beta   = (const float*)d_in[3];   // (1,1,1,CH)

    float* out      = (float*)d_out;
    float* new_xyz  = out;                             // B*S*3
    float* out_np   = out + (size_t)B_ * S_ * 3;       // B*S*K*128

    int* fps_idx = (int*)d_ws;                         // B*S ints
    int* knn_idx = fps_idx + (size_t)B_ * S_;          // B*S*K ints

    fps_kernel<<<B_, 1024, 0, stream>>>(xyz, new_xyz, fps_idx);

    const size_t dshm = (size_t)16 * N_ * sizeof(float);   // 256 KB LDS panel
    knn_kernel<<<B_ * (S_ / 16), 512, dshm, stream>>>(xyz, new_xyz, knn_idx);

    group_kernel<<<(B_ * S_) / 4, 256, 0, stream>>>(points, alpha, beta,
                                                    fps_idx, knn_idx, out_np);
}